// GNNModel_29600914604111
// MI455X (gfx1250) — compile-verified
//
#include <hip/hip_runtime.h>

typedef __attribute__((ext_vector_type(2))) float v2f;
typedef __attribute__((ext_vector_type(8))) float v8f;

#define WAVE 32

__device__ __forceinline__ void atomic_add_f32(float* p, float v) {
  __hip_atomic_fetch_add(p, v, __ATOMIC_RELAXED, __HIP_MEMORY_SCOPE_AGENT);
}

// ---------------- degree / dinv ----------------
__global__ void k_deg_init(float* __restrict__ deg, int n) {
  int i = blockIdx.x * blockDim.x + threadIdx.x;
  if (i < n) deg[i] = 1.0f;  // self loop
}

__global__ void k_deg_count(const int* __restrict__ dst, float* __restrict__ deg, int E) {
  int e = blockIdx.x * blockDim.x + threadIdx.x;
  if (e < E) atomic_add_f32(&deg[dst[e]], 1.0f);
}

__global__ void k_deg_rsqrt(float* __restrict__ deg, int n) {
  int i = blockIdx.x * blockDim.x + threadIdx.x;
  if (i < n) deg[i] = rsqrtf(deg[i]);
}

// ---------------- fp32 WMMA GEMM ----------------
// H = X(MxK) @ W(KxN)   (raw, needed for edge messages)
// AGG = H * dinv^2 + bias  (self-loop + bias init for the scatter-add)
// One wave per 16-row strip; N/16 column tiles; K stepped by 4 with
// V_WMMA_F32_16X16X4_F32.
// W is staged in LDS pair-interleaved: Ws[(k/2)*2N + 2n + {0,1}] = W[k][n], W[k+1][n]
// so each B fragment is a single aligned ds_load_b64 into an even VGPR pair.
template <int K, int N>
__global__ __launch_bounds__(256) void k_gemm_wmma(
    const float* __restrict__ X, const float* __restrict__ W,
    const float* __restrict__ bias, const float* __restrict__ dinv,
    float* __restrict__ H, float* __restrict__ AGG, int nStrips) {
  __shared__ float Ws[K * N];
  for (int i = threadIdx.x; i < (K / 2) * N; i += blockDim.x) {
    const int k2 = i / N;  // pair index (k = 2*k2)
    const int n  = i % N;
    v2f p;
    p.x = W[(2 * k2) * N + n];
    p.y = W[(2 * k2 + 1) * N + n];
    *(v2f*)&Ws[(size_t)k2 * 2 * N + 2 * n] = p;
  }
  __syncthreads();

  const int waveId = threadIdx.x / WAVE;
  const int strip = blockIdx.x * (blockDim.x / WAVE) + waveId;
  if (strip >= nStrips) return;  // wave-uniform: EXEC stays all-1s for WMMA

  const int lane = threadIdx.x & (WAVE - 1);
  const int half = lane >> 4;   // 0: K pair {0,1} / C rows 0-7 ; 1: K pair {2,3} / rows 8-15
  const int lr   = lane & 15;

  const float* xrow = X + (size_t)(strip * 16 + lr) * K;

  constexpr int NT = N / 16;
  v8f acc[NT];
#pragma unroll
  for (int t = 0; t < NT; ++t) acc[t] = {};

#pragma unroll 4
  for (int kk = 0; kk < K; kk += 4) {
    // A 16x4 fragment: lane row = lr, VGPR0/1 = K{0,1} (half 0) or K{2,3} (half 1)
    v2f a = *(const v2f*)(xrow + kk + 2 * half);
    // B 4x16 fragment: lane col = lr, same K split; pair-interleaved in LDS
    const float* wrow = &Ws[(size_t)(kk / 2 + half) * 2 * N];
#pragma unroll
    for (int t = 0; t < NT; ++t) {
      v2f b = *(const v2f*)(wrow + 2 * (t * 16 + lr));
      acc[t] = __builtin_amdgcn_wmma_f32_16x16x4_f32(
          false, a, false, b, (short)0, acc[t], false, false);
    }
  }

  // C/D layout: VGPR j -> row j (lanes 0-15) / row j+8 (lanes 16-31), col = lane%16
#pragma unroll
  for (int j = 0; j < 8; ++j) {
    const int orow = strip * 16 + j + 8 * half;
    const float di = dinv[orow];
    const float d2 = di * di;
    float* hrow = H + (size_t)orow * N;
    float* arow = AGG + (size_t)orow * N;
#pragma unroll
    for (int t = 0; t < NT; ++t) {
      const int col = t * 16 + lr;
      const float v = acc[t][j];
      hrow[col] = v;
      arow[col] = v * d2 + bias[col];
    }
  }
}

// ---------------- edge scatter-add ----------------
// thread = (edge, 4-channel quad): coalesced float4 gather of h[src],
// 4 hardware f32 atomics into AGG[dst].
template <int C>
__global__ void k_edge_agg(const int* __restrict__ src, const int* __restrict__ dst,
                           const float* __restrict__ dinv, const float* __restrict__ H,
                           float* __restrict__ AGG, int E) {
  constexpr int QP = C / 4;
  long long t = (long long)blockIdx.x * blockDim.x + threadIdx.x;
  if (t >= (long long)E * QP) return;
  const int e = (int)(t / QP);
  const int q = (int)(t % QP);
  const int s = src[e];
  const int d = dst[e];
  const float norm = dinv[s] * dinv[d];
  const float4 m = *(const float4*)(H + (size_t)s * C + q * 4);
  float* p = AGG + (size_t)d * C + q * 4;
  atomic_add_f32(p + 0, m.x * norm);
  atomic_add_f32(p + 1, m.y * norm);
  atomic_add_f32(p + 2, m.z * norm);
  atomic_add_f32(p + 3, m.w * norm);
}

__global__ void k_relu(float* __restrict__ a, int n) {
  int i = blockIdx.x * blockDim.x + threadIdx.x;
  if (i < n) a[i] = fmaxf(a[i], 0.0f);
}

extern "C" void kernel_launch(void* const* d_in, const int* in_sizes, int n_in,
                              void* d_out, int out_size, void* d_ws, size_t ws_size,
                              hipStream_t stream) {
  const float* x  = (const float*)d_in[0];
  const int*   ei = (const int*)d_in[1];
  const float* W1 = (const float*)d_in[2];
  const float* b1 = (const float*)d_in[3];
  const float* W2 = (const float*)d_in[4];
  const float* b2 = (const float*)d_in[5];

  constexpr int CIN = 128, CHID = 64, COUT = 32;
  const int N = in_sizes[0] / CIN;
  const int E = in_sizes[1] / 2;
  const int* src = ei;
  const int* dst = ei + E;

  // workspace layout (floats): dinv[N] | h1[N*64] | agg1[N*64] | h2[N*32]
  float* ws   = (float*)d_ws;
  float* dinv = ws;
  float* h1   = dinv + N;
  float* agg1 = h1 + (size_t)N * CHID;
  float* h2   = agg1 + (size_t)N * CHID;
  float* out  = (float*)d_out;  // agg2 lives directly in d_out

  const int B = 256;

  k_deg_init <<<(N + B - 1) / B, B, 0, stream>>>(dinv, N);
  k_deg_count<<<(E + B - 1) / B, B, 0, stream>>>(dst, dinv, E);
  k_deg_rsqrt<<<(N + B - 1) / B, B, 0, stream>>>(dinv, N);

  const int strips  = (N + 15) / 16;          // 6250 for N=100000 (exact)
  const int gblocks = (strips + 7) / 8;       // 8 waves per 256-thread block

  // layer 1: h1 = x @ W1 ; agg1 = h1*dinv^2 + b1
  k_gemm_wmma<CIN, CHID><<<gblocks, B, 0, stream>>>(x, W1, b1, dinv, h1, agg1, strips);

  const long long t1 = (long long)E * (CHID / 4);
  k_edge_agg<CHID><<<(unsigned)((t1 + B - 1) / B), B, 0, stream>>>(src, dst, dinv, h1, agg1, E);

  const int nh = N * CHID;
  k_relu<<<(nh + B - 1) / B, B, 0, stream>>>(agg1, nh);

  // layer 2: h2 = relu(agg1) @ W2 ; out = h2*dinv^2 + b2 (written into d_out)
  k_gemm_wmma<CHID, COUT><<<gblocks, B, 0, stream>>>(agg1, W2, b2, dinv, h2, out, strips);

  const long long t2 = (long long)E * (COUT / 4);
  k_edge_agg<COUT><<<(unsigned)((t2 + B - 1) / B), B, 0, stream>>>(src, dst, dinv, h2, out, E);
}